// DT_GCN_Lite_50757923504233
// MI455X (gfx1250) — compile-verified
//
#include <hip/hip_runtime.h>
#include <hip/hip_bf16.h>

#define N_NODES 100000
#define N_EDGES 1000000
#define IN_CH   64
#define OUT_CH  128
#define LDS_STRIDE (IN_CH + 1)   // pad to avoid 16-way LDS bank conflicts

typedef __attribute__((ext_vector_type(2))) float v2f;
typedef __attribute__((ext_vector_type(8))) float v8f;

// ---------------------------------------------------------------------------
// Phase 1: per-(edge, channel) gather + scale + scatter-add.
// Thread t -> edge e = t/64, channel c = t%64.
// x[col*64 + c] reads and agg[row*64 + c] atomics are coalesced across the
// 64 consecutive lanes of an edge (2 waves). Index loads are redundant per
// edge but hit L0.
// ---------------------------------------------------------------------------
__global__ __launch_bounds__(256) void gcn_scatter_kernel(
    const float* __restrict__ x,
    const long long* __restrict__ edge_index,   // [2, E] int64
    const float* __restrict__ edge_weight,
    float* __restrict__ agg)
{
    const long long t = (long long)blockIdx.x * blockDim.x + threadIdx.x;
    const long long total = (long long)N_EDGES * IN_CH;
    if (t >= total) return;

    const int e = (int)(t >> 6);          // t / 64
    const int c = (int)(t & 63);          // t % 64

    const long long row = edge_index[e];              // edge_index[0][e]
    const long long col = edge_index[(long long)N_EDGES + e]; // edge_index[1][e]
    const float w = edge_weight[e];

    const float v = w * x[col * IN_CH + c];
    atomicAdd(&agg[row * IN_CH + c], v);
}

// ---------------------------------------------------------------------------
// Phase 2: out = agg @ W^T + b via V_WMMA_F32_16X16X4_F32.
// One wave computes a 16(M) x 128(N) strip: 8 tiles of 16x16 f32 accumulators.
// K loop: 64 in steps of 4. W staged in padded LDS.
// ---------------------------------------------------------------------------
__global__ __launch_bounds__(256) void gcn_gemm_wmma_kernel(
    const float* __restrict__ agg,   // [N_NODES, IN_CH]
    const float* __restrict__ W,     // [OUT_CH, IN_CH] row-major; B = W^T
    const float* __restrict__ bias,  // [OUT_CH]
    float* __restrict__ out)         // [N_NODES, OUT_CH]
{
    __shared__ float Wlds[OUT_CH * LDS_STRIDE];  // 128*65*4 = 33,280 B

    // Cooperative W -> LDS copy (float4 granularity), row-padded.
    for (int i = threadIdx.x; i < (OUT_CH * IN_CH) / 4; i += 256) {
        const int r = (i * 4) / IN_CH;
        const int c = (i * 4) % IN_CH;
        const float4 w4 = ((const float4*)W)[i];
        float* dst = &Wlds[r * LDS_STRIDE + c];
        dst[0] = w4.x; dst[1] = w4.y; dst[2] = w4.z; dst[3] = w4.w;
    }
    __syncthreads();

    const int lane  = threadIdx.x & 31;
    const int wave  = (blockIdx.x * (256 / 32)) + (threadIdx.x >> 5);
    const int mBase = wave * 16;
    if (mBase >= N_NODES) return;            // wave-uniform: EXEC stays all-1s

    const int half = lane >> 4;              // 0: k-pair {0,1}; 1: k-pair {2,3}
    const int mn   = lane & 15;              // m for A, n_local for B/C/D

    v8f acc[8];
#pragma unroll
    for (int tIdx = 0; tIdx < 8; ++tIdx)
#pragma unroll
        for (int v = 0; v < 8; ++v) acc[tIdx][v] = 0.0f;

    const float* aRow = agg + (size_t)(mBase + mn) * IN_CH;

#pragma unroll
    for (int k = 0; k < IN_CH; k += 4) {
        const int k0 = k + half * 2;
        // A fragment: 16x4 f32 -> float2 per lane
        const float2 av = *(const float2*)(aRow + k0);
        v2f a; a.x = av.x; a.y = av.y;

#pragma unroll
        for (int tIdx = 0; tIdx < 8; ++tIdx) {
            const int n = tIdx * 16 + mn;
            // B fragment: B[k][n] = W[n][k]; contiguous k-pair in W's row
            const float2 bv = *(const float2*)(&Wlds[n * LDS_STRIDE + k0]);
            v2f b; b.x = bv.x; b.y = bv.y;
            acc[tIdx] = __builtin_amdgcn_wmma_f32_16x16x4_f32(
                /*neg_a=*/false, a, /*neg_b=*/false, b,
                /*c_mod=*/(short)0, acc[tIdx],
                /*reuse_a=*/false, /*reuse_b=*/false);
        }
    }

    // D layout: VGPR v -> row (v + 8*half), col = mn within each 16x16 tile.
#pragma unroll
    for (int tIdx = 0; tIdx < 8; ++tIdx) {
        const int n  = tIdx * 16 + mn;
        const float bv = bias[n];
#pragma unroll
        for (int v = 0; v < 8; ++v) {
            const int m = mBase + v + half * 8;
            out[(size_t)m * OUT_CH + n] = acc[tIdx][v] + bv;
        }
    }
}

// ---------------------------------------------------------------------------
// Launcher
// ---------------------------------------------------------------------------
extern "C" void kernel_launch(void* const* d_in, const int* in_sizes, int n_in,
                              void* d_out, int out_size, void* d_ws, size_t ws_size,
                              hipStream_t stream) {
    const float*     x  = (const float*)d_in[0];       // [N_NODES, IN_CH]
    const long long* ei = (const long long*)d_in[1];   // [2, N_EDGES] int64
    const float*     ew = (const float*)d_in[2];       // [N_EDGES]
    const float*     W  = (const float*)d_in[3];       // [OUT_CH, IN_CH]
    const float*     b  = (const float*)d_in[4];       // [OUT_CH]
    float*           out = (float*)d_out;              // [N_NODES, OUT_CH]

    float* agg = (float*)d_ws;                         // [N_NODES, IN_CH] scratch
    const size_t aggBytes = (size_t)N_NODES * IN_CH * sizeof(float);

    // Zero the accumulation buffer every call (graph-capturable memset node).
    hipMemsetAsync(agg, 0, aggBytes, stream);

    // Phase 1: scatter-add messages.
    {
        const long long total = (long long)N_EDGES * IN_CH;   // 64M threads
        const int block = 256;
        const int grid  = (int)((total + block - 1) / block);
        gcn_scatter_kernel<<<grid, block, 0, stream>>>(x, ei, ew, agg);
    }

    // Phase 2: WMMA GEMM + bias.
    {
        const int wavesNeeded  = N_NODES / 16;                 // 6250 exactly
        const int wavesPerBlk  = 256 / 32;                     // 8
        const int grid = (wavesNeeded + wavesPerBlk - 1) / wavesPerBlk;
        gcn_gemm_wmma_kernel<<<grid, 256, 0, stream>>>(agg, W, b, out);
    }
}